// YOLOLoss_20186346291605
// MI455X (gfx1250) — compile-verified
//
#include <hip/hip_runtime.h>
#include <math.h>

#define NUM_CLASSES 80
#define NTGT 512
#define BS 16
#define NA 3

#define C0 307200   // 16*3*80*80
#define C1 76800    // 16*3*40*40
#define C2 19200    // 16*3*20*20
#define CT 403200
#define K2_BLOCKS 256
#define K2_THREADS 256

typedef float v2f __attribute__((ext_vector_type(2)));
typedef float v8f __attribute__((ext_vector_type(8)));

// jnp.logaddexp(0, x) = max(x,0) + log1p(exp(-|x|))
__device__ __forceinline__ float softplusf(float x) {
  return fmaxf(x, 0.0f) + log1pf(expf(-fabsf(x)));
}
__device__ __forceinline__ float sigmoidf_(float x) {
  return 1.0f / (1.0f + expf(-x));
}

// Full wave32 sum via two chained V_WMMA_F32_16X16X4_F32 with an all-ones B.
// A layout (16x4 f32): lanes0-15 {v0=K0,v1=K1}, lanes16-31 {v0=K2,v1=K3}.
// With a = {s, 0}: rowsum(M) = s[M] + s[M+16]; summing D's 8 VGPRs gives
// S_lo on lanes 0-15 and S_hi on lanes 16-31; second WMMA adds the halves
// and broadcasts the wave total to every lane. EXEC must be all-ones here.
__device__ __forceinline__ float wave_reduce_wmma(float s) {
  v2f a;  a.x = s;    a.y = 0.0f;
  v2f bo; bo.x = 1.0f; bo.y = 1.0f;
  v8f c = {};
  v8f d = __builtin_amdgcn_wmma_f32_16x16x4_f32(false, a, false, bo, (short)0, c, false, false);
  float t = d[0] + d[1] + d[2] + d[3] + d[4] + d[5] + d[6] + d[7];
  v2f a2; a2.x = t; a2.y = 0.0f;
  v8f d2 = __builtin_amdgcn_wmma_f32_16x16x4_f32(false, a2, false, bo, (short)0, c, false, false);
  return d2[0];
}

// ---------------- Kernel 1: build targets + masked-cell losses ----------------
// One block, 512 threads (one per target). Replicates JAX scatter-set
// ("last duplicate wins") via an O(N^2) dedup in LDS, then the winning
// thread gathers the 85 channels at its cell and accumulates:
//   lbox, lobj_corr = 0.5*sp - pconf (on top of global 0.5*sum sp), lcls.
// Deterministic LDS tree reduction -> ws[0..3].
__global__ void __launch_bounds__(NTGT) yolo_targets_kernel(
    const float* __restrict__ tgt,
    const float* __restrict__ p0, const float* __restrict__ p1,
    const float* __restrict__ p2,
    float* __restrict__ ws) {
  __shared__ int   s_cell[NTGT];
  __shared__ float s_red[NTGT];
  const int tid = threadIdx.x;

  const float b = tgt[tid * 6 + 0];
  const float c = tgt[tid * 6 + 1];
  const float x = tgt[tid * 6 + 2];
  const float y = tgt[tid * 6 + 3];
  const float w = tgt[tid * 6 + 4];
  const float h = tgt[tid * 6 + 5];
  const float ssum = b + c + x + y + w + h;
  const bool valid = (ssum != 0.0f) && (b < 16.0f) && (c < 80.0f)
      && (x >= 0.0f) && (x <= 1.0f) && (y >= 0.0f) && (y <= 1.0f)
      && (w > 0.0f) && (w <= 1.0f) && (h > 0.0f) && (h <= 1.0f);

  const float ANCW[3][3]  = {{10.f, 16.f, 33.f}, {30.f, 62.f, 59.f}, {116.f, 156.f, 373.f}};
  const float ANCHh[3][3] = {{13.f, 30.f, 23.f}, {61.f, 45.f, 119.f}, {90.f, 198.f, 326.f}};
  const int GS[3] = {80, 40, 20};
  const float* preds[3] = {p0, p1, p2};

  float lbox = 0.f, lobjc = 0.f, lcls = 0.f;

  for (int s = 0; s < 3; ++s) {
    const int   G  = GS[s];
    const float Gf = (float)G;
    const float gx = x * Gf, gy = y * Gf, gw = w * Gf, gh = h * Gf;
    int gi = (int)fminf(gx, Gf - 1.0f); gi = gi < 0 ? 0 : (gi > G - 1 ? G - 1 : gi);
    int gj = (int)fminf(gy, Gf - 1.0f); gj = gj < 0 ? 0 : (gj > G - 1 ? G - 1 : gj);
    const float strd = 640.0f / Gf;

    int best = 0; float bestr = -1.0f, awb = 0.f, ahb = 0.f;
    for (int a = 0; a < 3; ++a) {
      const float aw = ANCW[s][a] / strd;
      const float ah = ANCHh[s][a] / strd;
      const float inter = fminf(gw, aw) * fminf(gh, ah);
      const float uni   = gw * gh + aw * ah - inter + 1e-7f;
      const float r = inter / uni;
      if (r > bestr) { bestr = r; best = a; awb = aw; ahb = ah; }  // first-max wins
    }

    const int bi = (int)b;
    const int cell = ((bi * NA + best) * G + gj) * G + gi;
    const bool sc_ok = valid && (bi >= 0) && (bi < BS);
    s_cell[tid] = sc_ok ? cell : -1;
    __syncthreads();

    bool winner = sc_ok;
    if (sc_ok) {
      for (int j = tid + 1; j < NTGT; ++j) {
        if (s_cell[j] == cell) { winner = false; break; }
      }
    }
    if (winner) {
      const float* pp = preds[s] + (long long)cell * 85;
      const float tx = gx - (float)gi;
      const float ty = gy - (float)gj;
      const float tw = logf(gw / (awb + 1e-16f) + 1e-16f);
      const float th = logf(gh / (ahb + 1e-16f) + 1e-16f);
      const float dx = sigmoidf_(pp[0]) - tx;
      const float dy = sigmoidf_(pp[1]) - ty;
      const float dw = pp[2] - tw;
      const float dh = pp[3] - th;
      lbox += dx * dx + dy * dy + dw * dw + dh * dh;
      const float pc = pp[4];
      lobjc += 0.5f * softplusf(pc) - pc;  // masked lobj minus over-counted noobj share
      int ci = (int)c; ci = ci < 0 ? 0 : (ci > NUM_CLASSES - 1 ? NUM_CLASSES - 1 : ci);
      float scls = 0.f;
      for (int k = 0; k < NUM_CLASSES; ++k) scls += softplusf(pp[5 + k]);
      scls -= pp[5 + ci];  // one-hot target term
      lcls += scls;
    }
    __syncthreads();
  }

  float vals[4] = {lbox, lobjc, lcls, valid ? 1.0f : 0.0f};
  for (int v = 0; v < 4; ++v) {
    s_red[tid] = vals[v];
    __syncthreads();
    for (int off = NTGT / 2; off >= 1; off >>= 1) {
      if (tid < off) s_red[tid] += s_red[tid + off];
      __syncthreads();
    }
    if (tid == 0) ws[v] = s_red[0];
    __syncthreads();
  }
}

// ---------------- Kernel 2: sum softplus(pconf) over all 403,200 cells -------
// Reads only channel 4 of each cell (stride 85 floats). Wave-level reduction
// uses chained f32 WMMAs; per-block partials written deterministically.
__global__ void __launch_bounds__(K2_THREADS) yolo_obj_sum_kernel(
    const float* __restrict__ p0, const float* __restrict__ p1,
    const float* __restrict__ p2,
    float* __restrict__ partials) {
  const int gid = blockIdx.x * K2_THREADS + threadIdx.x;
  const int nthreads = K2_BLOCKS * K2_THREADS;
  float acc = 0.f;
  for (int i = gid; i < CT; i += nthreads) {
    const float* p; int j;
    if (i < C0)           { p = p0; j = i; }
    else if (i < C0 + C1) { p = p1; j = i - C0; }
    else                  { p = p2; j = i - C0 - C1; }
    acc += softplusf(p[(long long)j * 85 + 4]);
  }
  const float wsum = wave_reduce_wmma(acc);  // all threads reach here: EXEC all-ones
  __shared__ float sw[K2_THREADS / 32];
  const int lane = threadIdx.x & 31, wid = threadIdx.x >> 5;
  if (lane == 0) sw[wid] = wsum;
  __syncthreads();
  if (threadIdx.x == 0) {
    float s = 0.f;
    for (int i = 0; i < K2_THREADS / 32; ++i) s += sw[i];
    partials[blockIdx.x] = s;
  }
}

// ---------------- Kernel 3: deterministic finalize ---------------------------
__global__ void yolo_finalize_kernel(const float* __restrict__ ws,
                                     const float* __restrict__ partials,
                                     float* __restrict__ out) {
  if (threadIdx.x == 0 && blockIdx.x == 0) {
    float ssp = 0.f;
    for (int i = 0; i < K2_BLOCKS; ++i) ssp += partials[i];
    float nt = 3.0f * ws[3];          // same valid set at every scale
    nt = fmaxf(nt, 1.0f);
    const float lobj = 0.5f * ssp + ws[1];
    out[0] = ws[0] / nt + lobj / (float)CT + ws[2] / nt;
  }
}

extern "C" void kernel_launch(void* const* d_in, const int* in_sizes, int n_in,
                              void* d_out, int out_size, void* d_ws, size_t ws_size,
                              hipStream_t stream) {
  (void)in_sizes; (void)n_in; (void)out_size; (void)ws_size;
  const float* p0 = (const float*)d_in[0];
  const float* p1 = (const float*)d_in[1];
  const float* p2 = (const float*)d_in[2];
  const float* tg = (const float*)d_in[3];
  float* ws = (float*)d_ws;           // ws[0..3]: lbox, lobj_corr, lcls, valid_count
  float* partials = ws + 8;           // 256 block partials (~1 KB total)

  yolo_targets_kernel<<<1, NTGT, 0, stream>>>(tg, p0, p1, p2, ws);
  yolo_obj_sum_kernel<<<K2_BLOCKS, K2_THREADS, 0, stream>>>(p0, p1, p2, partials);
  yolo_finalize_kernel<<<1, 32, 0, stream>>>(ws, partials, (float*)d_out);
}